// Codebook_frosz_65618510348309
// MI455X (gfx1250) — compile-verified
//
#include <hip/hip_runtime.h>
#include <hip/hip_bf16.h>

typedef float v2f __attribute__((ext_vector_type(2)));
typedef float v8f __attribute__((ext_vector_type(8)));

#define KCODES 1024
#define DDIM   256
#define HWSZ   4096   // 64*64
#define NTOT   65536  // 16*64*64
#define ROWS   32     // rows (n) per workgroup: 2 M-tiles of 16

// ---------------------------------------------------------------------------
// Kernel 1: enorm[k] = ||emb[k]||^2
// ---------------------------------------------------------------------------
__global__ __launch_bounds__(128)
void vq_enorm_kernel(const float* __restrict__ emb, float* __restrict__ enorm) {
    int k = blockIdx.x * 128 + threadIdx.x;
    if (k < KCODES) {
        const float* e = emb + (size_t)k * DDIM;
        float s = 0.f;
        #pragma unroll 8
        for (int d = 0; d < DDIM; ++d) { float v = e[d]; s += v * v; }
        enorm[k] = s;
    }
}

// ---------------------------------------------------------------------------
// Kernel 2: fused  S = z_tile * emb^T (fp32 WMMA, 2x4 register blocking)
//           -> argmin_k(enorm - 2S) -> indices + gathered z_q.
// 128-thread (4-wave) workgroup per 32 consecutive NHWC rows.
// Each wave: 32 rows x its 256-code range; per d-step 2 ds + 4 glb -> 8 WMMA.
// ---------------------------------------------------------------------------
#define ASTRIDE 257   // odd stride: conflict-free across the 16 M-rows

__global__ __launch_bounds__(128)
void vq_main_kernel(const float* __restrict__ z,
                    const float* __restrict__ emb,
                    const float* __restrict__ enorm,
                    int*   __restrict__ idx_out,
                    float* __restrict__ zq_out) {
    __shared__ float lds_z[ROWS * ASTRIDE];   // [m][d]
    __shared__ float cand_d[4][ROWS][16];     // [wave][m][j]
    __shared__ int   cand_k[4][ROWS][16];
    __shared__ int   best_idx[ROWS];

    const int tid = threadIdx.x;
    const int n0  = blockIdx.x * ROWS;        // 32 rows: same b,h; w0..w0+31
    const size_t zb_off = (size_t)(n0 >> 12) * (DDIM * HWSZ) + (size_t)(n0 & (HWSZ - 1));
    const float* zb = z + zb_off;             // + d*HWSZ + m

    // ---- stage A-tile (32 x 256 fp32) into LDS, coalesced 128B rows ----
    for (int f = tid; f < ROWS * DDIM; f += 128) {
        int d = f >> 5, m = f & 31;
        lds_z[m * ASTRIDE + d] = zb[(size_t)d * HWSZ + m];
    }
    __syncthreads();

    const int lane  = tid & 31;        // wave32
    const int wave  = tid >> 5;        // 0..3 -> k range [wave*256, +256)
    const int j     = lane & 15;       // code offset inside 16-wide tile
    const int hi    = lane >> 4;       // 0/1 -> K-halves of the x4 fragment
    const int dsel  = hi * 2;          // fragment K offset within d-chunk
    const int mbase = hi * 8;          // C-layout M base for this half-wave

    float bestD0[8], bestD1[8];        // M-tile 0 (rows 0..15), tile 1 (16..31)
    int   bestK0[8], bestK1[8];
    #pragma unroll
    for (int r = 0; r < 8; ++r) {
        bestD0[r] = 3.4e38f; bestK0[r] = 0;
        bestD1[r] = 3.4e38f; bestK1[r] = 0;
    }

    const float* alds0 = &lds_z[(lane & 15) * ASTRIDE];       // rows 0..15
    const float* alds1 = alds0 + 16 * ASTRIDE;                // rows 16..31

    for (int g = 0; g < 4; ++g) {                 // 4 groups of 4 k-tiles
        const int k0 = wave * 256 + g * 64;       // tiles k0, +16, +32, +48
        const float* eb0 = emb + (size_t)(k0 + j) * DDIM;
        const float* eb1 = eb0 + (size_t)16 * DDIM;
        const float* eb2 = eb0 + (size_t)32 * DDIM;
        const float* eb3 = eb0 + (size_t)48 * DDIM;

        // prefetch next group's B rows (clamped in-range; speculative)
        {
            int kn = (wave * 256 + ((g + 1) & 3) * 64 + j) & (KCODES - 1);
            __builtin_prefetch(emb + (size_t)kn * DDIM, 0, 3);
        }

        v8f a00 = {0.f,0.f,0.f,0.f,0.f,0.f,0.f,0.f};   // acc[mtile][ktile]
        v8f a01 = a00, a02 = a00, a03 = a00;
        v8f a10 = a00, a11 = a00, a12 = a00, a13 = a00;

        #pragma unroll 4
        for (int d0 = 0; d0 < DDIM; d0 += 4) {
            v2f fa0, fa1, fb0, fb1, fb2, fb3;
            fa0.x = alds0[d0 + dsel]; fa0.y = alds0[d0 + dsel + 1];
            fa1.x = alds1[d0 + dsel]; fa1.y = alds1[d0 + dsel + 1];
            fb0.x = eb0[d0 + dsel];   fb0.y = eb0[d0 + dsel + 1];
            fb1.x = eb1[d0 + dsel];   fb1.y = eb1[d0 + dsel + 1];
            fb2.x = eb2[d0 + dsel];   fb2.y = eb2[d0 + dsel + 1];
            fb3.x = eb3[d0 + dsel];   fb3.y = eb3[d0 + dsel + 1];
            a00 = __builtin_amdgcn_wmma_f32_16x16x4_f32(false, fa0, false, fb0,
                      (short)0, a00, false, false);
            a01 = __builtin_amdgcn_wmma_f32_16x16x4_f32(false, fa0, false, fb1,
                      (short)0, a01, false, false);
            a02 = __builtin_amdgcn_wmma_f32_16x16x4_f32(false, fa0, false, fb2,
                      (short)0, a02, false, false);
            a03 = __builtin_amdgcn_wmma_f32_16x16x4_f32(false, fa0, false, fb3,
                      (short)0, a03, false, false);
            a10 = __builtin_amdgcn_wmma_f32_16x16x4_f32(false, fa1, false, fb0,
                      (short)0, a10, false, false);
            a11 = __builtin_amdgcn_wmma_f32_16x16x4_f32(false, fa1, false, fb1,
                      (short)0, a11, false, false);
            a12 = __builtin_amdgcn_wmma_f32_16x16x4_f32(false, fa1, false, fb2,
                      (short)0, a12, false, false);
            a13 = __builtin_amdgcn_wmma_f32_16x16x4_f32(false, fa1, false, fb3,
                      (short)0, a13, false, false);
        }

        // distance (up to row-constant ||z||^2): enorm[k] - 2*S[m][k]
        const float en0 = enorm[k0 + j];
        const float en1 = enorm[k0 + 16 + j];
        const float en2 = enorm[k0 + 32 + j];
        const float en3 = enorm[k0 + 48 + j];
        #pragma unroll
        for (int r = 0; r < 8; ++r) {
            // k-tiles folded in ascending-k order: preserves first-min semantics
            float d0v = en0 - 2.0f * a00[r];
            if (d0v < bestD0[r]) { bestD0[r] = d0v; bestK0[r] = k0 + j; }
            float d1v = en1 - 2.0f * a01[r];
            if (d1v < bestD0[r]) { bestD0[r] = d1v; bestK0[r] = k0 + 16 + j; }
            float d2v = en2 - 2.0f * a02[r];
            if (d2v < bestD0[r]) { bestD0[r] = d2v; bestK0[r] = k0 + 32 + j; }
            float d3v = en3 - 2.0f * a03[r];
            if (d3v < bestD0[r]) { bestD0[r] = d3v; bestK0[r] = k0 + 48 + j; }

            float e0v = en0 - 2.0f * a10[r];
            if (e0v < bestD1[r]) { bestD1[r] = e0v; bestK1[r] = k0 + j; }
            float e1v = en1 - 2.0f * a11[r];
            if (e1v < bestD1[r]) { bestD1[r] = e1v; bestK1[r] = k0 + 16 + j; }
            float e2v = en2 - 2.0f * a12[r];
            if (e2v < bestD1[r]) { bestD1[r] = e2v; bestK1[r] = k0 + 32 + j; }
            float e3v = en3 - 2.0f * a13[r];
            if (e3v < bestD1[r]) { bestD1[r] = e3v; bestK1[r] = k0 + 48 + j; }
        }
    }

    #pragma unroll
    for (int r = 0; r < 8; ++r) {
        int m = mbase + r;
        cand_d[wave][m][j]      = bestD0[r];
        cand_k[wave][m][j]      = bestK0[r];
        cand_d[wave][m + 16][j] = bestD1[r];
        cand_k[wave][m + 16][j] = bestK1[r];
    }
    __syncthreads();

    // ---- per-row reduction over 4 waves x 16 lanes; first-min tie-break ----
    if (tid < ROWS) {
        int m = tid;
        float bd = 3.4e38f; int bk = KCODES;
        for (int w = 0; w < 4; ++w)
            for (int jj = 0; jj < 16; ++jj) {
                float dd = cand_d[w][m][jj];
                int   kk = cand_k[w][m][jj];
                if (dd < bd || (dd == bd && kk < bk)) { bd = dd; bk = kk; }
            }
        best_idx[m] = bk;
        idx_out[n0 + m] = bk;
    }
    __syncthreads();

    // ---- gather z_q = emb[idx] back to NCHW (coalesced 128B rows) ----
    float* zq_b = zq_out + zb_off;
    for (int f = tid; f < ROWS * DDIM; f += 128) {
        int d = f >> 5, m = f & 31;
        zq_b[(size_t)d * HWSZ + m] = emb[(size_t)best_idx[m] * DDIM + d];
    }
}

// ---------------------------------------------------------------------------
extern "C" void kernel_launch(void* const* d_in, const int* in_sizes, int n_in,
                              void* d_out, int out_size, void* d_ws, size_t ws_size,
                              hipStream_t stream) {
    const float* z   = (const float*)d_in[0];   // [16,256,64,64] fp32
    const float* emb = (const float*)d_in[1];   // [1024,256] fp32

    float* enorm = (float*)d_ws;                // 4 KB scratch
    int*   idx   = (int*)d_out;                 // first N elements: indices
    float* zq    = (float*)d_out + NTOT;        // then 16*256*64*64 floats

    vq_enorm_kernel<<<(KCODES + 127) / 128, 128, 0, stream>>>(emb, enorm);
    vq_main_kernel<<<NTOT / ROWS, 128, 0, stream>>>(z, emb, enorm, idx, zq);
}